// GATConv_62199716381234
// MI455X (gfx1250) — compile-verified
//
#include <hip/hip_runtime.h>
#include <hip/hip_bf16.h>
#include <math.h>

#define N_NODES 50000
#define N_EDGES 800000
#define NODE_F 256
#define EDGE_F 8
#define HEADS 4
#define OUTF 64
#define HF (HEADS * OUTF)       // 256
#define MTILES (N_NODES / 16)   // 3125, exact
#define NTILES (HF / 16)        // 16
#define NT_GRP 8                // ntiles per wave (register blocking)

typedef float v2f __attribute__((ext_vector_type(2)));
typedef float v8f __attribute__((ext_vector_type(8)));

// ---------------------------------------------------------------------------
// Kernel 1: node projections via exact-fp32 WMMA, register-blocked.
// One wave owns a 16x128 output strip (8 N-tiles) of one projection:
//   - A fragment loaded ONCE per k-step, reused by 8 WMMAs (8x A reuse)
//   - 8 independent accumulators -> no WMMA->WMMA RAW hazards
//   - waves within a block share (which, group): B loads are L1 broadcasts
//   which==0 : src_fc = feat @ W_src^T                  -> ws
//   which==1 : d_out  = feat @ W_dst^T + b_dst          (residual base)
// ---------------------------------------------------------------------------
__global__ void gat_nodeproj_wmma(const float* __restrict__ feat,
                                  const float* __restrict__ Wsrc,
                                  const float* __restrict__ Wdst,
                                  const float* __restrict__ bdst,
                                  float* __restrict__ src_fc,
                                  float* __restrict__ out_rst) {
    const int wave = (blockIdx.x * blockDim.x + threadIdx.x) >> 5;
    const int lane = threadIdx.x & 31;
    const int JOBS = 2 * (NTILES / NT_GRP) * MTILES;   // 12500
    if (wave >= JOBS) return;

    // mt varies fastest so the 8 waves of a block share (which, g).
    const int mt    = wave % MTILES;
    const int wg    = wave / MTILES;       // 0..3
    const int g     = wg & 1;              // ntile group (0 or 1)
    const int which = wg >> 1;             // 0=src proj, 1=dst proj

    const float* __restrict__ W = which ? Wdst : Wsrc;  // [256 out][256 in] row-major
    const int l15  = lane & 15;
    const int koff = (lane >> 4) * 2;      // lanes 16-31 hold K+2,K+3 (ISA 7.12.2)

    const float* __restrict__ arow = feat + (size_t)(mt * 16 + l15) * NODE_F;
    const float* __restrict__ bcol[NT_GRP];
#pragma unroll
    for (int t = 0; t < NT_GRP; ++t)       // B[k][col] = W[col][k] (row-major W)
        bcol[t] = W + (size_t)((g * NT_GRP + t) * 16 + l15) * NODE_F;

    v8f acc[NT_GRP] = {};
#pragma unroll 2
    for (int k = 0; k < NODE_F; k += 4) {
        const v2f a = *(const v2f*)(arow + k + koff);
#pragma unroll
        for (int t = 0; t < NT_GRP; ++t) {
            const v2f b = *(const v2f*)(bcol[t] + k + koff);
            acc[t] = __builtin_amdgcn_wmma_f32_16x16x4_f32(
                         /*neg_a=*/false, a, /*neg_b=*/false, b,
                         /*c_mod=*/(short)0, acc[t],
                         /*reuse_a=*/false, /*reuse_b=*/false);
        }
    }

    // D layout: VGPR v -> M = v + 8*(lane>=16), N = lane%16
    const int mbase = mt * 16 + ((lane >> 4) << 3);
    float* __restrict__ outp = which ? out_rst : src_fc;
#pragma unroll
    for (int t = 0; t < NT_GRP; ++t) {
        const int nidx  = (g * NT_GRP + t) * 16 + l15;
        const float bias = which ? bdst[nidx] : 0.0f;
#pragma unroll
        for (int v = 0; v < 8; ++v)
            outp[(size_t)(mbase + v) * HF + nidx] = acc[t][v] + bias;
    }
}

// ---------------------------------------------------------------------------
// Kernel 2: attn_src = feat @ W_as^T, attn_dst = feat @ W_ad^T  ([N,4] each).
// One wave per node; lane-strided coalesced reads + shuffle reduction.
// ---------------------------------------------------------------------------
__global__ void gat_node_attn(const float* __restrict__ feat,
                              const float* __restrict__ Was,
                              const float* __restrict__ Wad,
                              float* __restrict__ attn_src,
                              float* __restrict__ attn_dst) {
    const int node = (blockIdx.x * blockDim.x + threadIdx.x) >> 5;
    const int lane = threadIdx.x & 31;
    if (node >= N_NODES) return;
    const float* __restrict__ f = feat + (size_t)node * NODE_F;
    float fv[8];
#pragma unroll
    for (int j = 0; j < 8; ++j) fv[j] = f[lane + 32 * j];

#pragma unroll
    for (int h = 0; h < HEADS; ++h) {
        const float* ws_ = Was + h * NODE_F;
        const float* wd_ = Wad + h * NODE_F;
        float ss = 0.0f, sd = 0.0f;
#pragma unroll
        for (int j = 0; j < 8; ++j) {
            ss += fv[j] * ws_[lane + 32 * j];
            sd += fv[j] * wd_[lane + 32 * j];
        }
        for (int off = 16; off > 0; off >>= 1) {
            ss += __shfl_xor(ss, off, 32);
            sd += __shfl_xor(sd, off, 32);
        }
        if (lane == 0) {
            attn_src[node * HEADS + h] = ss;
            attn_dst[node * HEADS + h] = sd;
        }
    }
}

// Order-preserving float->uint mapping for atomic max on floats.
__device__ __forceinline__ unsigned fkey(float v) {
    unsigned b = __float_as_uint(v);
    return (b & 0x80000000u) ? ~b : (b | 0x80000000u);
}
__device__ __forceinline__ float funkey(unsigned k) {
    return __uint_as_float((k & 0x80000000u) ? (k ^ 0x80000000u) : ~k);
}

// ---------------------------------------------------------------------------
// Kernel 3: per-edge attention logits + leaky relu + segment-max.
// ---------------------------------------------------------------------------
__global__ void gat_edge_logits(const float* __restrict__ attn_src,
                                const float* __restrict__ attn_dst,
                                const float* __restrict__ feat_edge,
                                const float* __restrict__ Wae,
                                const int* __restrict__ src,
                                const int* __restrict__ dst,
                                float* __restrict__ ebuf,
                                unsigned* __restrict__ maxbuf) {
    const int i = blockIdx.x * blockDim.x + threadIdx.x;
    if (i >= N_EDGES) return;
    const int s = src[i], d = dst[i];
    const float4 as = *(const float4*)(attn_src + (size_t)s * HEADS);
    const float4 ad = *(const float4*)(attn_dst + (size_t)d * HEADS);
    float fe[EDGE_F];
#pragma unroll
    for (int j = 0; j < EDGE_F; ++j) fe[j] = feat_edge[(size_t)i * EDGE_F + j];

    float e[HEADS];
    const float* asp = (const float*)&as;
    const float* adp = (const float*)&ad;
#pragma unroll
    for (int h = 0; h < HEADS; ++h) {
        float ae = 0.0f;
#pragma unroll
        for (int j = 0; j < EDGE_F; ++j) ae += fe[j] * Wae[h * EDGE_F + j];
        float v = asp[h] + adp[h] + ae;
        v = (v >= 0.0f) ? v : 0.2f * v;           // leaky relu
        e[h] = v;
        atomicMax(&maxbuf[(size_t)d * HEADS + h], fkey(v));
    }
    *(float4*)(ebuf + (size_t)i * HEADS) = make_float4(e[0], e[1], e[2], e[3]);
}

// ---------------------------------------------------------------------------
// Kernel 4: ex = exp(e - max[dst]); denom[dst] += ex   (in-place on ebuf).
// ---------------------------------------------------------------------------
__global__ void gat_edge_exp(float* __restrict__ ebuf,
                             const unsigned* __restrict__ maxbuf,
                             const int* __restrict__ dst,
                             float* __restrict__ denom) {
    const int i = blockIdx.x * blockDim.x + threadIdx.x;
    if (i >= N_EDGES) return;
    const int d = dst[i];
    float4 e = *(const float4*)(ebuf + (size_t)i * HEADS);
    float* ep = (float*)&e;
#pragma unroll
    for (int h = 0; h < HEADS; ++h) {
        const float m  = funkey(maxbuf[(size_t)d * HEADS + h]);
        const float ex = expf(ep[h] - m);
        ep[h] = ex;
        atomicAdd(&denom[(size_t)d * HEADS + h], ex);
    }
    *(float4*)(ebuf + (size_t)i * HEADS) = e;
}

// ---------------------------------------------------------------------------
// Kernel 5: out[dst] += src_fc[src] * (ex / denom[dst]), one wave per edge.
// Lane L handles feature slots [8L, 8L+8) => head h = L>>3. L2-resident.
// ---------------------------------------------------------------------------
__global__ void gat_edge_aggregate(const float* __restrict__ src_fc,
                                   const float* __restrict__ exbuf,
                                   const float* __restrict__ denom,
                                   const int* __restrict__ src,
                                   const int* __restrict__ dst,
                                   float* __restrict__ out) {
    const int edge = (blockIdx.x * blockDim.x + threadIdx.x) >> 5;
    const int lane = threadIdx.x & 31;
    if (edge >= N_EDGES) return;
    const int s = src[edge], d = dst[edge];
    const int h = lane >> 3;
    const float a = exbuf[(size_t)edge * HEADS + h] /
                    denom[(size_t)d * HEADS + h];

    const float4* __restrict__ p =
        (const float4*)(src_fc + (size_t)s * HF + lane * 8);
    const float4 v0 = p[0];
    const float4 v1 = p[1];
    float* __restrict__ o = out + (size_t)d * HF + lane * 8;
    atomicAdd(o + 0, v0.x * a);
    atomicAdd(o + 1, v0.y * a);
    atomicAdd(o + 2, v0.z * a);
    atomicAdd(o + 3, v0.w * a);
    atomicAdd(o + 4, v1.x * a);
    atomicAdd(o + 5, v1.y * a);
    atomicAdd(o + 6, v1.z * a);
    atomicAdd(o + 7, v1.w * a);
}

// ---------------------------------------------------------------------------
extern "C" void kernel_launch(void* const* d_in, const int* in_sizes, int n_in,
                              void* d_out, int out_size, void* d_ws, size_t ws_size,
                              hipStream_t stream) {
    const float* feat      = (const float*)d_in[0];
    const float* feat_edge = (const float*)d_in[1];
    const float* Wsrc      = (const float*)d_in[2];
    const float* Wdst      = (const float*)d_in[3];
    const float* bdst      = (const float*)d_in[4];
    const float* Was       = (const float*)d_in[5];
    const float* Wad       = (const float*)d_in[6];
    const float* Wae       = (const float*)d_in[7];
    const int*   src       = (const int*)d_in[8];
    const int*   dst       = (const int*)d_in[9];
    float*       out       = (float*)d_out;

    // Workspace carve-up (all 16B-aligned): ~67 MB total.
    float*    src_fc = (float*)d_ws;                                // N*HF
    float*    attn_s = src_fc + (size_t)N_NODES * HF;               // N*H
    float*    attn_d = attn_s + (size_t)N_NODES * HEADS;            // N*H
    float*    ebuf   = attn_d + (size_t)N_NODES * HEADS;            // E*H
    unsigned* maxb   = (unsigned*)(ebuf + (size_t)N_EDGES * HEADS); // N*H
    float*    denom  = (float*)(maxb + (size_t)N_NODES * HEADS);    // N*H

    // Per-call re-init (graph-capture safe).
    hipMemsetAsync(maxb,  0, (size_t)N_NODES * HEADS * sizeof(unsigned), stream);
    hipMemsetAsync(denom, 0, (size_t)N_NODES * HEADS * sizeof(float),    stream);

    // 1) WMMA projections: 12500 waves, each a 16x128 strip; 8 waves/block.
    {
        const int waves  = 2 * (NTILES / NT_GRP) * MTILES;  // 12500
        const int blocks = (waves + 7) / 8;                 // 1563
        gat_nodeproj_wmma<<<blocks, 256, 0, stream>>>(feat, Wsrc, Wdst, bdst,
                                                      src_fc, out);
    }
    // 2) Node-level attention scalars: 1 wave per node.
    gat_node_attn<<<(N_NODES + 7) / 8, 256, 0, stream>>>(feat, Was, Wad,
                                                         attn_s, attn_d);
    // 3) Edge logits + segment max.
    gat_edge_logits<<<(N_EDGES + 255) / 256, 256, 0, stream>>>(
        attn_s, attn_d, feat_edge, Wae, src, dst, ebuf, maxb);
    // 4) exp + denom.
    gat_edge_exp<<<(N_EDGES + 255) / 256, 256, 0, stream>>>(ebuf, maxb, dst, denom);
    // 5) weighted scatter-add on top of residual already in `out`.
    gat_edge_aggregate<<<(N_EDGES + 7) / 8, 256, 0, stream>>>(src_fc, ebuf, denom,
                                                              src, dst, out);
}